// LegacyElboGenerativeModelTop_39977555591238
// MI455X (gfx1250) — compile-verified
//
#include <hip/hip_runtime.h>
#include <math.h>

// Problem constants (from reference): T=256 steps, BS=512 batch, N=2048 features.
#define T_STEPS 256
#define BATCH   512
#define NFEAT   2048
#define WAVES_PER_BLOCK 8
#define BLOCK_THREADS   256          // 8 wave32s
#define RPT 64                       // 2048 / 32 lanes: regs per lane per feature-vector

// ---------------- CDNA5 async global->LDS helpers --------------------------
#if defined(__gfx1250__) && __has_builtin(__builtin_amdgcn_global_load_async_to_lds_b32)
#define HAVE_ASYNC_LDS 1
#else
#define HAVE_ASYNC_LDS 0
#endif

typedef __attribute__((address_space(1))) int GINT;  // global (AS1) int*
typedef __attribute__((address_space(3))) int LINT;  // LDS (AS3) int*

__device__ __forceinline__ void async_copy_b32(const float* gsrc, float* ldst) {
#if HAVE_ASYNC_LDS
  __builtin_amdgcn_global_load_async_to_lds_b32((GINT*)const_cast<float*>(gsrc),
                                                (LINT*)ldst, 0, 0);
#else
  *ldst = *gsrc;
#endif
}

__device__ __forceinline__ void async_wait_all() {
#if HAVE_ASYNC_LDS
#if __has_builtin(__builtin_amdgcn_s_wait_asynccnt)
  __builtin_amdgcn_s_wait_asynccnt(0);
#else
  asm volatile("s_wait_asynccnt 0" ::: "memory");
#endif
#endif
}

// ---------------- Main kernel: register-resident recurrent scan ------------
// One wave32 per batch element (512 waves). All feature-dim vectors live in
// VGPRs (5 x 64 regs: biases, w_inq, w_in, w_out, h). Noise + LPF tables are
// staged into LDS via CDNA5 async global->LDS ops.
__global__ __launch_bounds__(BLOCK_THREADS, 1)
void elbo_scan_kernel(const float* __restrict__ noises,    // [T, BS]
                      const float* __restrict__ z_biases,  // [N]
                      const float* __restrict__ w_in,      // [N]
                      const float* __restrict__ w_inq,     // [N]
                      const float* __restrict__ sq,        // [T] (ws)
                      const float* __restrict__ yl,        // [T] (ws)
                      const float* __restrict__ p_llr,
                      const float* __restrict__ p_llrd,
                      const float* __restrict__ p_sigb,
                      const float* __restrict__ p_os,
                      const float* __restrict__ p_ufs,
                      const float* __restrict__ p_lwd,
                      float* __restrict__ out)             // [T, BS]
{
  __shared__ float  nbuf[T_STEPS * WAVES_PER_BLOCK];    // noise[t][wave]  8 KB
  __shared__ float2 sybuf[T_STEPS];                     // {sq[t], yl[t]}  2 KB

  const int tid  = threadIdx.x;
  const int lane = tid & 31;
  const int w    = tid >> 5;
  const int k0   = blockIdx.x * WAVES_PER_BLOCK;
  const int k    = k0 + w;

  const float sigb = p_sigb[0];
  const float lr0  = expf(p_llr[0]);
  const float lrd  = expf(p_llrd[0]);
  const float wd   = expf(p_lwd[0]);
  const float os   = p_os[0];
  const float ufs  = p_ufs[0];

  // Stage this block's noise slice [T][8] via async global->LDS (CDNA5 path).
  for (int idx = tid; idx < T_STEPS * WAVES_PER_BLOCK; idx += BLOCK_THREADS) {
    const int t  = idx >> 3;
    const int ww = idx & 7;
    async_copy_b32(&noises[t * BATCH + k0 + ww], &nbuf[idx]);
  }
  // Stage the scalar LPF tables (tid covers exactly T_STEPS).
  async_copy_b32(&sq[tid], &sybuf[tid].x);
  async_copy_b32(&yl[tid], &sybuf[tid].y);

  // Per-wave register state: feature dim striped across 32 lanes x 64 regs.
  float bia[RPT], wq[RPT], win[RPT], wout[RPT], h[RPT];
#pragma unroll
  for (int r = 0; r < RPT; ++r) {
    const int j = r * 32 + lane;
    bia[r]  = sigb * z_biases[j];
    wq[r]   = w_inq[j];
    win[r]  = w_in[j];
    wout[r] = 0.0f;
  }

  async_wait_all();
  __syncthreads();

  float u = 0.0f, e = 0.0f, lrm = 1.0f;

#pragma unroll 1
  for (int t = 0; t < T_STEPS; ++t) {
    const float2 sy  = sybuf[t];
    const float  s   = sy.x;                 // q_scale * qlp_t
    const float  ylp = sy.y;                 // ylp_t
    const float  x   = fmaf(u, ufs, e) + nbuf[t * WAVES_PER_BLOCK + w];

    // 4 accumulators per sum to break the serial FMA dependency chains.
    float d0 = 0.f, d1 = 0.f, d2 = 0.f, d3 = 0.f;
    float q0 = 0.f, q1 = 0.f, q2 = 0.f, q3 = 0.f;
#pragma unroll
    for (int r = 0; r < RPT; r += 4) {
      const float h0 = fmaxf(fmaf(x, win[r+0], fmaf(s, wq[r+0], bia[r+0])), 0.f);
      const float h1 = fmaxf(fmaf(x, win[r+1], fmaf(s, wq[r+1], bia[r+1])), 0.f);
      const float h2 = fmaxf(fmaf(x, win[r+2], fmaf(s, wq[r+2], bia[r+2])), 0.f);
      const float h3 = fmaxf(fmaf(x, win[r+3], fmaf(s, wq[r+3], bia[r+3])), 0.f);
      h[r+0] = h0; h[r+1] = h1; h[r+2] = h2; h[r+3] = h3;
      d0 = fmaf(wout[r+0], h0, d0);
      d1 = fmaf(wout[r+1], h1, d1);
      d2 = fmaf(wout[r+2], h2, d2);
      d3 = fmaf(wout[r+3], h3, d3);
      q0 = fmaf(h0, h0, q0);
      q1 = fmaf(h1, h1, q1);
      q2 = fmaf(h2, h2, q2);
      q3 = fmaf(h3, h3, q3);
    }
    float dot = (d0 + d1) + (d2 + d3);
    float hsq = (q0 + q1) + (q2 + q3);

    // In-wave butterfly reduction (wave32), both sums together.
#pragma unroll
    for (int m = 16; m > 0; m >>= 1) {
      dot += __shfl_xor(dot, m, 32);
      hsq += __shfl_xor(hsq, m, 32);
    }
    u = dot;
    if (lane == 0) out[t * BATCH + k] = os * u;

    e = ylp - u;
    const float c2  = e * (lr0 * lrm);                 // e * lr
    const float nrm = sqrtf(1e-4f + c2 * c2 * hsq);    // ||dw|| = |e*lr| * ||h||
    lrm *= expf(-lrd * nrm);

#pragma unroll
    for (int r = 0; r < RPT; ++r)
      wout[r] = wd * fmaf(c2, h[r], wout[r]);          // w_out = wd*(w_out + dw)
  }
}

// ---------------- Kernel 2: scalar LPF pre-scan (batch-independent) --------
// sq[t] = q_scale * qlp_t ;  yl[t] = ylp_t   (post-update values at step t)
__global__ void lpf_prescan_kernel(const float* __restrict__ ys,
                                   const float* __restrict__ qs,
                                   const float* __restrict__ p_qscale,
                                   const float* __restrict__ p_tauq_m1,
                                   const float* __restrict__ p_tauy_m1,
                                   float* __restrict__ sq,
                                   float* __restrict__ yl) {
  if (threadIdx.x != 0 || blockIdx.x != 0) return;
  const float tauq = 1.0f + log1pf(expf(p_tauq_m1[0]));
  const float tauy = 1.0f + log1pf(expf(p_tauy_m1[0]));
  const float qsc  = p_qscale[0];
  float qlp = 0.0f, ylp = 0.0f;
  for (int t = 0; t < T_STEPS; ++t) {
    qlp = (1.0f - tauq) * qlp + tauq * qs[t];
    ylp = (1.0f - tauy) * ylp + tauy * ys[t];
    sq[t] = qsc * qlp;
    yl[t] = ylp;
  }
}

// ---------------- Host-side launcher ---------------------------------------
extern "C" void kernel_launch(void* const* d_in, const int* in_sizes, int n_in,
                              void* d_out, int out_size, void* d_ws, size_t ws_size,
                              hipStream_t stream) {
  (void)in_sizes; (void)n_in; (void)out_size; (void)ws_size;

  const float* noises   = (const float*)d_in[0];
  const float* ys       = (const float*)d_in[1];
  const float* qs       = (const float*)d_in[2];
  const float* z_biases = (const float*)d_in[3];
  const float* w_in     = (const float*)d_in[4];
  const float* w_inq    = (const float*)d_in[5];
  const float* llr      = (const float*)d_in[6];
  const float* llrd     = (const float*)d_in[7];
  const float* sigb     = (const float*)d_in[8];
  const float* osc      = (const float*)d_in[9];
  const float* ufs      = (const float*)d_in[10];
  const float* lwd      = (const float*)d_in[11];
  const float* qsc      = (const float*)d_in[12];
  const float* tq       = (const float*)d_in[13];
  const float* ty       = (const float*)d_in[14];

  float* out = (float*)d_out;
  float* sq  = (float*)d_ws;          // [T]
  float* ylb = sq + T_STEPS;          // [T]

  lpf_prescan_kernel<<<1, 64, 0, stream>>>(ys, qs, qsc, tq, ty, sq, ylb);
  elbo_scan_kernel<<<BATCH / WAVES_PER_BLOCK, BLOCK_THREADS, 0, stream>>>(
      noises, z_biases, w_in, w_inq, sq, ylb,
      llr, llrd, sigb, osc, ufs, lwd, out);
}